// GroupLoss_30863634989288
// MI455X (gfx1250) — compile-verified
//
#include <hip/hip_runtime.h>
#include <hip/hip_bf16.h>

typedef float v2f __attribute__((ext_vector_type(2)));
typedef float v8f __attribute__((ext_vector_type(8)));

#define NFLAT 128   // 16*8
#define NPERS 10
#define NKP   17
#define HMAP  128
#define WMAP  128

// One wave32 per n. Lane p (< NPERS) handles person p: gather 17 embeds,
// per-person mean/variance (two-pass, matching reference numerics), then
// cross-lane pairwise hinge. Writes per-n partials: ws[n]=within_n,
// ws[128+n]=across_n.
__global__ __launch_bounds__(32)
void group_loss_per_n(const float* __restrict__ preds,
                      const float* __restrict__ gt,
                      float* __restrict__ ws) {
  const int n    = blockIdx.x;   // 0..127
  const int lane = threadIdx.x;  // 0..31
  constexpr float INV_STRIDE = 1.0f / 4.0f;  // 1/FEAT_STRIDE

  float vals[NKP];
  unsigned vmask = 0u;
  float sum = 0.0f, cnt = 0.0f;

  if (lane < NPERS) {
    const float* g  = gt + (size_t)(n * NPERS + lane) * NKP * 2;
    const float* pk = preds + (size_t)n * NKP * HMAP * WMAP;
#pragma unroll
    for (int k = 0; k < NKP; ++k) {
      float gx = g[2 * k + 0];
      float gy = g[2 * k + 1];
      // jnp.round == round-half-to-even == v_rndne
      int x = (int)rintf(gx * INV_STRIDE);
      int y = (int)rintf(gy * INV_STRIDE);
      bool valid = (x >= 0) && (x < WMAP) && (y >= 0) && (y < HMAP);
      float v = 0.0f;
      if (valid) {
        v = pk[(size_t)k * HMAP * WMAP + y * WMAP + x];
        vmask |= (1u << k);
        sum += v;
        cnt += 1.0f;
      }
      vals[k] = v;
    }
  }

  float safe = fmaxf(cnt, 1.0f);
  float mean = sum / safe;   // person_embed (0 when cnt==0, matches reference)

  float wsum = 0.0f;
  if (lane < NPERS) {
#pragma unroll
    for (int k = 0; k < NKP; ++k) {
      if (vmask & (1u << k)) {
        float d = vals[k] - mean;
        wsum += d * d;
      }
    }
  }
  float within_p = (cnt > 0.0f) ? (wsum / safe) : 0.0f;

  float pv = (lane < NPERS && cnt > 0.0f) ? 1.0f : 0.0f;
  float e  = (lane < NPERS) ? mean : 0.0f;

  // pairwise hinge among valid persons (off-diagonal), lane p sums its row
  float hinge = 0.0f;
#pragma unroll
  for (int j = 0; j < NPERS; ++j) {
    float ej  = __shfl(e, j, 32);
    float pvj = __shfl(pv, j, 32);
    if (j != lane) {
      float h = fmaxf(1.0f - fabsf(e - ej), 0.0f);
      hinge += h * pv * pvj;
    }
  }

  // wave32 reductions: hinge sum, valid-person count S, within sum
  float hs = hinge, ps = pv, wsum_all = within_p;
#pragma unroll
  for (int off = 16; off; off >>= 1) {
    hs       += __shfl_xor(hs, off, 32);
    ps       += __shfl_xor(ps, off, 32);
    wsum_all += __shfl_xor(wsum_all, off, 32);
  }

  if (lane == 0) {
    float within_n = wsum_all / (float)NPERS;
    // pair.sum = S^2 - S  (pvalid is 0/1)
    float denom = ps * ps - ps;
    float across_n = (denom > 0.0f) ? (hs / fmaxf(denom, 1.0f)) : 0.0f;
    ws[n]         = within_n;
    ws[NFLAT + n] = across_n;
  }
}

// Final reduction of 2x128 partials using the CDNA5 matrix pipe as a
// cross-lane reducer. With A == all-ones(16x4):
//   D[m,n] = sum_k B[k,n]  (column sums; every row of D identical)
// so summing c[0] over all 32 lanes (rows 0 and 8 of D) = 2 * grand total,
// independent of B's exact lane<->element mapping. Two accumulated WMMAs
// cover all 128 values per output.
__global__ __launch_bounds__(32)
void group_loss_reduce(const float* __restrict__ ws, float* __restrict__ out) {
  const int lane = threadIdx.x;  // 0..31, full EXEC (WMMA requirement)

  v2f ones;
  ones.x = 1.0f;
  ones.y = 1.0f;

#pragma unroll
  for (int t = 0; t < 2; ++t) {          // t=0: within, t=1: across
    const float* src = ws + t * NFLAT;
    v2f b0, b1;
    b0.x = src[lane];
    b0.y = src[32 + lane];
    b1.x = src[64 + lane];
    b1.y = src[96 + lane];

    v8f c = {};
    c = __builtin_amdgcn_wmma_f32_16x16x4_f32(
        /*neg_a=*/false, ones, /*neg_b=*/false, b0,
        /*c_mod=*/(short)0, c, /*reuse_a=*/false, /*reuse_b=*/false);
    c = __builtin_amdgcn_wmma_f32_16x16x4_f32(
        /*neg_a=*/false, ones, /*neg_b=*/false, b1,
        /*c_mod=*/(short)0, c, /*reuse_a=*/false, /*reuse_b=*/false);

    float s = c[0];                       // D[0,lane] / D[8,lane-16]
#pragma unroll
    for (int off = 16; off; off >>= 1) s += __shfl_xor(s, off, 32);
    // s == 2 * sum of all 128 partials; mean over N=128, weight 1.0
    if (lane == 0) out[t] = s * (1.0f / (2.0f * (float)NFLAT));
  }
}

extern "C" void kernel_launch(void* const* d_in, const int* in_sizes, int n_in,
                              void* d_out, int out_size, void* d_ws, size_t ws_size,
                              hipStream_t stream) {
  const float* preds = (const float*)d_in[0];  // (16,8,17,128,128) f32
  const float* gt    = (const float*)d_in[1];  // (16,8,10,17,2)   f32
  float* ws  = (float*)d_ws;                   // 256 floats used
  float* out = (float*)d_out;                  // [within, across]

  group_loss_per_n<<<NFLAT, 32, 0, stream>>>(preds, gt, ws);
  group_loss_reduce<<<1, 32, 0, stream>>>(ws, out);
}